// ConvolutionModule_79259326480930
// MI455X (gfx1250) — compile-verified
//
#include <hip/hip_runtime.h>

#define D 128

typedef float v2f __attribute__((ext_vector_type(2)));
typedef float v8f __attribute__((ext_vector_type(8)));

// ---------------- degree / inverse degree ----------------

__global__ void sage_degree_kernel(const int* __restrict__ dst, float* __restrict__ deg, int E) {
  int e = blockIdx.x * blockDim.x + threadIdx.x;
  if (e < E) unsafeAtomicAdd(&deg[dst[e]], 1.0f);
}

__global__ void sage_invdeg_kernel(const float* __restrict__ deg, float* __restrict__ invd, int n) {
  int i = blockIdx.x * blockDim.x + threadIdx.x;
  if (i < n) invd[i] = 1.0f / fmaxf(deg[i], 1.0f);
}

// ---------------- edge scatter: agg[dst] += h[src] ----------------
// one wave per edge; lane l moves bytes [l*16, l*16+16) of the 512B feature row.

__global__ void sage_scatter_kernel(const float* __restrict__ h, const int* __restrict__ src,
                                    const int* __restrict__ dst, float* __restrict__ agg, int E) {
  int e = blockIdx.x * 8 + (threadIdx.x >> 5);
  int lane = threadIdx.x & 31;
  if (e >= E) return;
  int s = src[e];
  int d = dst[e];
  const float4 v = ((const float4*)(h + (size_t)s * D))[lane];
  float* a = agg + (size_t)d * D + lane * 4;
  unsafeAtomicAdd(a + 0, v.x);
  unsafeAtomicAdd(a + 1, v.y);
  unsafeAtomicAdd(a + 2, v.z);
  unsafeAtomicAdd(a + 3, v.w);
}

// ---------------- fused dual-GEMM + bias + ReLU ----------------
// out[m, :] = relu(h[m,:] @ Ws^T + (agg[m,:]*invd[m]) @ Wn^T + b)
// Block = 256 threads = 8 waves. Block handles 16 rows (m0..m0+15); wave w
// handles cols [16w, 16w+16). Requires nrows % 16 == 0 (50000 = 3125*16).
// A (16x4 f32) lane layout: lanes 0-15 row=lane K={k0,k0+1}; lanes 16-31 same
// rows, K={k0+2,k0+3}.  B (4x16) mirrored with N across lanes.

__global__ void __launch_bounds__(256)
sage_gemm_kernel(const float* __restrict__ h, const float* __restrict__ agg,
                 const float* __restrict__ invd,
                 const float* __restrict__ Ws, const float* __restrict__ Wn,
                 const float* __restrict__ bias, float* __restrict__ out) {
  __shared__ float sA[16][132];  // h rows, padded stride vs 64 banks
  __shared__ float sG[16][132];  // inv_deg-scaled agg rows

  const int t = threadIdx.x;
  const int m0 = blockIdx.x * 16;

  // stage 16x128 panels of h and scaled agg into LDS (float4 per thread x2)
  const float* hrow = h + (size_t)m0 * D;
  const float* grow = agg + (size_t)m0 * D;
#pragma unroll
  for (int rep = 0; rep < 2; ++rep) {
    int idx = rep * 1024 + t * 4;
    int r = idx >> 7;
    int c = idx & 127;
    float4 v = *(const float4*)(hrow + r * D + c);
    *(float4*)(&sA[r][c]) = v;
    float id = invd[m0 + r];
    float4 g = *(const float4*)(grow + r * D + c);
    g.x *= id; g.y *= id; g.z *= id; g.w *= id;
    *(float4*)(&sG[r][c]) = g;
  }
  __syncthreads();

  const int lane = t & 31;
  const int wave = t >> 5;
  const int lid  = lane & 15;
  const int half = lane >> 4;
  const int n0   = wave * 16;

  const float* wsrow = Ws + (size_t)(n0 + lid) * D;  // B[k,n] = W[n,k]
  const float* wnrow = Wn + (size_t)(n0 + lid) * D;

  v8f acc = {0.f, 0.f, 0.f, 0.f, 0.f, 0.f, 0.f, 0.f};

#pragma unroll
  for (int k0 = 0; k0 < D; k0 += 4) {
    const int ka = k0 + 2 * half;
    v2f a  = { sA[lid][ka], sA[lid][ka + 1] };
    v2f bs = { wsrow[ka],   wsrow[ka + 1] };
    acc = __builtin_amdgcn_wmma_f32_16x16x4_f32(false, a, false, bs, (short)0, acc,
                                                false, false);
    v2f g  = { sG[lid][ka], sG[lid][ka + 1] };
    v2f bn = { wnrow[ka],   wnrow[ka + 1] };
    acc = __builtin_amdgcn_wmma_f32_16x16x4_f32(false, g, false, bn, (short)0, acc,
                                                false, false);
  }

  // C/D layout: VGPR i -> row m0 + i + 8*half, col n0 + lid
  const int col = n0 + lid;
  const float bv = bias[col];
#pragma unroll
  for (int i = 0; i < 8; ++i) {
    int row = m0 + 8 * half + i;
    out[(size_t)row * D + col] = fmaxf(acc[i] + bv, 0.0f);
  }
}

// ---------------- driver ----------------

extern "C" void kernel_launch(void* const* d_in, const int* in_sizes, int n_in,
                              void* d_out, int out_size, void* d_ws, size_t ws_size,
                              hipStream_t stream) {
  const float* node = (const float*)d_in[0];
  const int*   src  = (const int*)d_in[1];
  const int*   dst  = (const int*)d_in[2];
  const float* Ws[3] = {(const float*)d_in[3], (const float*)d_in[6], (const float*)d_in[9]};
  const float* Wn[3] = {(const float*)d_in[4], (const float*)d_in[7], (const float*)d_in[10]};
  const float* bb[3] = {(const float*)d_in[5], (const float*)d_in[8], (const float*)d_in[11]};

  const int N = in_sizes[0] / D;
  const int E = in_sizes[1];

  float* hA   = (float*)d_ws;
  float* hB   = hA + (size_t)N * D;
  float* agg  = hB + (size_t)N * D;
  float* invd = agg + (size_t)N * D;
  float* deg  = invd + N;

  hipMemsetAsync(deg, 0, (size_t)N * sizeof(float), stream);
  sage_degree_kernel<<<(E + 255) / 256, 256, 0, stream>>>(dst, deg, E);
  sage_invdeg_kernel<<<(N + 255) / 256, 256, 0, stream>>>(deg, invd, N);

  const float* hcur = node;
  float* houts[3] = {hA, hB, (float*)d_out};
  const int rowTiles = N / 16;  // N = 50000 = 3125 * 16

  for (int l = 0; l < 3; ++l) {
    hipMemsetAsync(agg, 0, (size_t)N * D * sizeof(float), stream);
    sage_scatter_kernel<<<(E + 7) / 8, 256, 0, stream>>>(hcur, src, dst, agg, E);
    sage_gemm_kernel<<<rowTiles, 256, 0, stream>>>(hcur, agg, invd, Ws[l], Wn[l], bb[l],
                                                   houts[l]);
    hcur = houts[l];
  }
}